// SimplifiedTGN_17540646437558
// MI455X (gfx1250) — compile-verified
//
#include <hip/hip_runtime.h>

#define HDIM 64
#define NDIM 128
#define EDIM 64
// LDS row strides padded so stride % 64 == 4: a half-wave reading
// consecutive rows at the same column pair covers all 64 banks (no conflicts).
#define S64 68    // padded stride for 64-wide rows
#define S128 132  // padded stride for 128-wide rows

typedef __attribute__((ext_vector_type(2))) float v2f;
typedef __attribute__((ext_vector_type(8))) float v8f;

__device__ __forceinline__ v2f ld2(const float* p) { return *(const v2f*)p; }

__device__ __forceinline__ v8f wmma_f32_k4(v2f a, v2f b, v8f c) {
  // D = A(16x4) * B(4x16) + C, full fp32 precision
  return __builtin_amdgcn_wmma_f32_16x16x4_f32(false, a, false, b, (short)0, c,
                                               false, false);
}

__global__ void zero_u32_kernel(unsigned* __restrict__ p, long n) {
  long i = (long)blockIdx.x * blockDim.x + threadIdx.x;
  long stride = (long)gridDim.x * blockDim.x;
  for (; i < n; i += stride) p[i] = 0u;
}

__global__ void mark_needed_kernel(const int* __restrict__ post_idx,
                                   int* __restrict__ needed, int P) {
  int i = blockIdx.x * blockDim.x + threadIdx.x;
  if (i < P) needed[post_idx[i]] = 1;
}

// node_emb = relu(node_features @ W_node^T + b_node); W_node staged in LDS.
__global__ void node_emb_kernel(const float* __restrict__ nf,
                                const float* __restrict__ Wn,
                                const float* __restrict__ bn,
                                float* __restrict__ emb, int N) {
  __shared__ float sW[64 * S128];  // 33.8 KB
  for (int i = threadIdx.x; i < 64 * NDIM; i += blockDim.x)
    sW[(i >> 7) * S128 + (i & 127)] = Wn[i];
  __syncthreads();

  const int lane = threadIdx.x & 31;
  const int wv = threadIdx.x >> 5;
  const int mi = lane & 15;
  const int half = lane >> 4;
  const int kb = half << 1;

  float bnv[4];
#pragma unroll
  for (int nt = 0; nt < 4; ++nt) bnv[nt] = bn[nt * 16 + mi];

  const int numTiles = (N + 15) >> 4;
  const int wstride = gridDim.x * (blockDim.x >> 5);
  for (int tile = blockIdx.x * (blockDim.x >> 5) + wv; tile < numTiles;
       tile += wstride) {  // wave-uniform loop, EXEC all-ones at WMMA
    int row = tile * 16 + mi;
    if (row > N - 1) row = N - 1;

    v8f acc[4];
#pragma unroll
    for (int nt = 0; nt < 4; ++nt) {
      float bv = bnv[nt];
      v8f t = {bv, bv, bv, bv, bv, bv, bv, bv};
      acc[nt] = t;
    }

    const float* arow = nf + (long)row * NDIM;
#pragma unroll 4
    for (int k = 0; k < NDIM; k += 4) {
      v2f a = ld2(arow + k + kb);  // A[16x4]: lane=M, lane[4] selects K pair
#pragma unroll
      for (int nt = 0; nt < 4; ++nt) {
        v2f b = ld2(&sW[(nt * 16 + mi) * S128 + k + kb]);  // conflict-free LDS
        acc[nt] = wmma_f32_k4(a, b, acc[nt]);
      }
    }

    float* erow = emb + (long)(tile * 16) * HDIM;
    if (tile * 16 + 16 <= N) {
      // fast path: full tile, no per-element exec masking
#pragma unroll
      for (int nt = 0; nt < 4; ++nt)
#pragma unroll
        for (int v = 0; v < 8; ++v) {
          int mr = v + 8 * half;
          float x = acc[nt][v];
          erow[mr * HDIM + nt * 16 + mi] = x > 0.f ? x : 0.f;
        }
    } else {
#pragma unroll
      for (int nt = 0; nt < 4; ++nt)
#pragma unroll
        for (int v = 0; v < 8; ++v) {
          int mr = v + 8 * half;
          if (tile * 16 + mr < N) {
            float x = acc[nt][v];
            erow[mr * HDIM + nt * 16 + mi] = x > 0.f ? x : 0.f;
          }
        }
    }
  }
}

// Stream-compact edge ids whose dst is a needed (post) node.
__global__ void compact_edges_kernel(const int* __restrict__ edge_dst,
                                     const int* __restrict__ needed,
                                     int* __restrict__ edge_list,
                                     int* __restrict__ counter, int E) {
  int i = blockIdx.x * blockDim.x + threadIdx.x;
  int pred = 0;
  if (i < E) pred = needed[edge_dst[i]];
  unsigned mask = __builtin_amdgcn_ballot_w32(pred != 0);
  int lane = threadIdx.x & 31;
  int total = __builtin_popcount(mask);
  int base = 0;
  if (lane == 0 && total > 0) base = atomicAdd(counter, total);
  base = __builtin_amdgcn_readfirstlane(base);
  if (pred) {
    int prefix = __builtin_popcount(mask & ((1u << lane) - 1u));
    edge_list[base + prefix] = i;
  }
}

// Fused per-edge pipeline over compacted edges (tiles of 16):
//   eemb = relu(EA[e] @ We^T + be)
//   msg  = node_emb[src] @ Wc[:, :64]^T + eemb @ Wc[:, 64:]^T + bc
//   messages[dst] += msg   (float atomics, L2-resident buffer)
// Weights live in LDS (staged once per block); waves grid-stride over tiles.
__global__ void edge_msg_kernel(const float* __restrict__ node_emb,
                                const float* __restrict__ edge_attr,
                                const int* __restrict__ edge_src,
                                const int* __restrict__ edge_dst,
                                const int* __restrict__ edge_list,
                                const int* __restrict__ counter,
                                const float* __restrict__ We,
                                const float* __restrict__ be,
                                const float* __restrict__ Wc,
                                const float* __restrict__ bc,
                                float* __restrict__ messages) {
  __shared__ float sWe[64 * S64];        // 17.4 KB
  __shared__ float sWc[64 * S128];       // 33.8 KB
  __shared__ float sTile[8 * 16 * S64];  // 34.8 KB (4.35 KB per wave)

  for (int i = threadIdx.x; i < 64 * EDIM; i += blockDim.x)
    sWe[(i >> 6) * S64 + (i & 63)] = We[i];
  for (int i = threadIdx.x; i < 64 * (2 * HDIM); i += blockDim.x)
    sWc[(i >> 7) * S128 + (i & 127)] = Wc[i];
  __syncthreads();  // single block barrier, before any divergent wave work

  const int lane = threadIdx.x & 31;
  const int wv = threadIdx.x >> 5;
  const int mi = lane & 15;
  const int half = lane >> 4;
  const int kb = half << 1;
  float* L = &sTile[wv * 16 * S64];

  float bev[4], bcv[4];
#pragma unroll
  for (int nt = 0; nt < 4; ++nt) {
    bev[nt] = be[nt * 16 + mi];
    bcv[nt] = bc[nt * 16 + mi];
  }

  const int count = counter[0];
  const int numTiles = (count + 15) >> 4;
  const int wstride = gridDim.x * (blockDim.x >> 5);
  for (int tile = blockIdx.x * (blockDim.x >> 5) + wv; tile < numTiles;
       tile += wstride) {  // wave-uniform loop
    __builtin_prefetch(&edge_list[(long)(tile + wstride) * 16], 0, 1);

    int eslot = tile * 16 + mi;
    if (eslot > count - 1) eslot = count - 1;  // clamp partial tile; compute
    const int eid = edge_list[eslot];          // garbage rows, mask the stores
    const int srcn = edge_src[eid];

    // ---- GEMM1: edge embedding ----
    v8f acc1[4];
#pragma unroll
    for (int nt = 0; nt < 4; ++nt) {
      float bv = bev[nt];
      v8f t = {bv, bv, bv, bv, bv, bv, bv, bv};
      acc1[nt] = t;
    }
    const float* arow = edge_attr + (long)eid * EDIM;
#pragma unroll 4
    for (int k = 0; k < EDIM; k += 4) {
      v2f a = ld2(arow + k + kb);
#pragma unroll
      for (int nt = 0; nt < 4; ++nt) {
        v2f b = ld2(&sWe[(nt * 16 + mi) * S64 + k + kb]);
        acc1[nt] = wmma_f32_k4(a, b, acc1[nt]);
      }
    }

    // relu + restage C-layout -> row-major LDS (padded, conflict-free)
#pragma unroll
    for (int nt = 0; nt < 4; ++nt)
#pragma unroll
      for (int v = 0; v < 8; ++v) {
        int mr = v + 8 * half;
        float x = acc1[nt][v];
        L[mr * S64 + nt * 16 + mi] = x > 0.f ? x : 0.f;
      }
    // same-wave DS ops are in-order; fence compiler + hw before cross-lane read
    asm volatile("s_wait_dscnt 0" ::: "memory");

    // ---- GEMM2: message = node_term + edge_term + b_conv ----
    v8f acc2[4];
#pragma unroll
    for (int nt = 0; nt < 4; ++nt) {
      float bv = bcv[nt];
      v8f t = {bv, bv, bv, bv, bv, bv, bv, bv};
      acc2[nt] = t;
    }
    const float* nrow = node_emb + (long)srcn * HDIM;
#pragma unroll 4
    for (int k = 0; k < HDIM; k += 4) {
      v2f a = ld2(nrow + k + kb);  // gathered row; node_emb is L2-resident
#pragma unroll
      for (int nt = 0; nt < 4; ++nt) {
        v2f b = ld2(&sWc[(nt * 16 + mi) * S128 + k + kb]);  // Wc_node
        acc2[nt] = wmma_f32_k4(a, b, acc2[nt]);
      }
    }
#pragma unroll 4
    for (int k = 0; k < HDIM; k += 4) {
      v2f a = ld2(&L[mi * S64 + k + kb]);  // A from padded LDS tile
#pragma unroll
      for (int nt = 0; nt < 4; ++nt) {
        v2f b = ld2(&sWc[(nt * 16 + mi) * S128 + HDIM + k + kb]);  // Wc_edge
        acc2[nt] = wmma_f32_k4(a, b, acc2[nt]);
      }
    }

    // ---- scatter-add into messages[dst] ----
    int nvalid = count - tile * 16;
    if (nvalid >= 16) {
      // fast path: full tile, unguarded atomics (no exec-mask churn)
#pragma unroll
      for (int v = 0; v < 8; ++v) {
        int mr = v + 8 * half;
        int e2 = edge_list[tile * 16 + mr];
        int d = edge_dst[e2];
        float* mrow = messages + (long)d * HDIM;
#pragma unroll
        for (int nt = 0; nt < 4; ++nt)
          atomicAdd(mrow + nt * 16 + mi, acc2[nt][v]);
      }
    } else {
      // last partial tile only
#pragma unroll
      for (int v = 0; v < 8; ++v) {
        int mr = v + 8 * half;
        if (mr < nvalid) {
          int e2 = edge_list[tile * 16 + mr];
          int d = edge_dst[e2];
          float* mrow = messages + (long)d * HDIM;
#pragma unroll
          for (int nt = 0; nt < 4; ++nt)
            atomicAdd(mrow + nt * 16 + mi, acc2[nt][v]);
        }
      }
    }
  }
}

// logits -> sigmoid at the P post nodes; one wave per post row.
__global__ void out_kernel(const float* __restrict__ node_emb,
                           const float* __restrict__ messages,
                           const int* __restrict__ post_idx,
                           const float* __restrict__ Wout,
                           const float* __restrict__ bout,
                           float* __restrict__ out, int P) {
  const int lane = threadIdx.x & 31;
  const int p = blockIdx.x * (blockDim.x >> 5) + (threadIdx.x >> 5);
  if (p >= P) return;
  const long node = post_idx[p];
  const float* e = node_emb + node * HDIM;
  const float* ms = messages + node * HDIM;
  float s = 0.f;
#pragma unroll
  for (int h0 = 0; h0 < HDIM; h0 += 32) {
    int h = h0 + lane;
    s += (e[h] + ms[h]) * Wout[h];
  }
#pragma unroll
  for (int off = 16; off > 0; off >>= 1) s += __shfl_down(s, off, 32);
  if (lane == 0) {
    float z = s + bout[0];
    out[p] = 1.f / (1.f + __expf(-z));
  }
}

extern "C" void kernel_launch(void* const* d_in, const int* in_sizes, int n_in,
                              void* d_out, int out_size, void* d_ws,
                              size_t ws_size, hipStream_t stream) {
  const float* node_features = (const float*)d_in[0];
  const int* edge_index = (const int*)d_in[1];
  const float* edge_attr = (const float*)d_in[2];
  const int* post_idx = (const int*)d_in[3];
  const float* W_node = (const float*)d_in[4];
  const float* b_node = (const float*)d_in[5];
  const float* W_edge = (const float*)d_in[6];
  const float* b_edge = (const float*)d_in[7];
  const float* W_conv = (const float*)d_in[8];
  const float* b_conv = (const float*)d_in[9];
  const float* W_out = (const float*)d_in[10];
  const float* b_out = (const float*)d_in[11];

  const int N = in_sizes[0] / NDIM;
  const int E = in_sizes[1] / 2;
  const int P = in_sizes[3];

  // workspace layout (contiguous): node_emb | messages | needed | counter | edge_list
  float* node_emb = (float*)d_ws;
  float* messages = node_emb + (long)N * HDIM;
  int* needed = (int*)(messages + (long)N * HDIM);
  int* counter = needed + N;
  int* edge_list = counter + 1;

  const int* edge_src = edge_index;
  const int* edge_dst = edge_index + E;

  // zero messages + needed + counter in one pass (they're contiguous)
  long nzero = (long)N * HDIM + N + 1;
  zero_u32_kernel<<<2048, 256, 0, stream>>>((unsigned*)messages, nzero);

  mark_needed_kernel<<<(P + 255) / 256, 256, 0, stream>>>(post_idx, needed, P);

  int nodeTiles = (N + 15) / 16;
  int nodeBlocks = (nodeTiles + 7) / 8;
  if (nodeBlocks > 512) nodeBlocks = 512;  // grid-stride, amortize LDS staging
  node_emb_kernel<<<nodeBlocks, 256, 0, stream>>>(node_features, W_node, b_node,
                                                  node_emb, N);

  compact_edges_kernel<<<(E + 255) / 256, 256, 0, stream>>>(edge_dst, needed,
                                                            edge_list, counter, E);

  // persistent-style grid; waves self-limit by reading the device-side count
  int maxTiles = (E + 15) / 16;
  int edgeBlocks = (maxTiles + 7) / 8;
  if (edgeBlocks > 1024) edgeBlocks = 1024;
  edge_msg_kernel<<<edgeBlocks, 256, 0, stream>>>(
      node_emb, edge_attr, edge_src, edge_dst, edge_list, counter, W_edge,
      b_edge, W_conv, b_conv, messages);

  out_kernel<<<(P + 7) / 8, 256, 0, stream>>>(node_emb, messages, post_idx,
                                              W_out, b_out, (float*)d_out, P);
}